// SimplePatchScorer_83562883711077
// MI455X (gfx1250) — compile-verified
//
#include <hip/hip_runtime.h>

// v_wmma_f32_16x16x4_f32 operand fragments (wave32)
typedef float v2f __attribute__((ext_vector_type(2)));
typedef float v4f __attribute__((ext_vector_type(4)));
typedef float v8f __attribute__((ext_vector_type(8)));

#define ROWS_PER_GROUP 49
#define ROW_STRIDE     772                         // 768 + 4-float pad (bank-conflict free)
#define LDSW_OFF       (ROWS_PER_GROUP * ROW_STRIDE)   // 37828 floats
#define LDS_FLOATS     (LDSW_OFF + 768)                // 38596 floats = 154384 B

__global__ void patch_scorer_wmma(const float* __restrict__ x,
                                  const float* __restrict__ W,
                                  const float* __restrict__ bias,
                                  float* __restrict__ out)
{
    extern __shared__ float lds[];
    const int tid = threadIdx.x;
    const int b   = blockIdx.x >> 2;   // image
    const int g   = blockIdx.x & 3;    // ph-group: ph in [4g, 4g+4)

    // ---- Stage 0: stage W (768 floats) into LDS ----
    if (tid < 192) {
        v4f wv = *(const v4f*)(W + tid * 4);
        *(v4f*)(&lds[LDSW_OFF + tid * 4]) = wv;
    }

    // ---- Stage 1: coalesced read of the 168 image rows this group needs,
    //      scatter into LDS in the scrambled row-major (j,k) order. ----
    const float* xb = x + (size_t)b * 150528;   // 3*224*224
    for (int it = 0; it < 37; ++it) {
        int idx4 = tid + it * 256;              // 9408 float4 loads total
        if (idx4 < 9408) {
            int w4 = idx4 % 56;                 // column/4
            int t  = idx4 / 56;
            int ph = t & 3;
            int t2 = t >> 2;
            int hn = t2 % 14;
            int c  = t2 / 14;
            int h  = hn * 16 + g * 4 + ph;
            v4f v = *(const v4f*)(xb + c * 50176 + h * 224 + w4 * 4);
            int wbase = w4 * 4;
            #pragma unroll
            for (int e = 0; e < 4; ++e) {
                int w  = wbase + e;             // pw never crosses a 16-boundary here
                int wn = w >> 4, pw = w & 15;
                // f' = local flat index inside this group's 37632-element span
                int f = (ph * 16 + pw) * 588 + c * 196 + hn * 14 + wn;
                int j = (int)((unsigned)f / 768u);   // local output row 0..48
                int k = f - j * 768;
                lds[j * ROW_STRIDE + k] = v[e];
            }
        }
    }
    __syncthreads();

    // ---- Stage 2: GEMV via v_wmma_f32_16x16x4_f32, 4 waves x 16-row tiles ----
    const int wave = tid >> 5;
    if (wave >= 4) return;
    const int lane = tid & 31;
    const int half = lane >> 4;        // selects K+2 side of A/B fragments
    const int m    = lane & 15;        // A-matrix row within tile
    const int s    = min(wave * 16, 33);   // tile starts 0,16,32,33 -> cover rows 0..48

    const float b0 = bias[0];
    v8f acc;
    #pragma unroll
    for (int i = 0; i < 8; ++i) acc[i] = b0;   // C init = bias, accumulated once

    const float* arow = &lds[(s + m) * ROW_STRIDE + 2 * half];
    const float* wrow = &lds[LDSW_OFF + 2 * half];
    for (int k0 = 0; k0 < 768; k0 += 4) {
        v2f a  = *(const v2f*)(arow + k0);     // A[m][k0 + 2*half + {0,1}]
        v2f bw = *(const v2f*)(wrow + k0);     // B rows K=2*half+{0,1}, all 16 cols = W
        acc = __builtin_amdgcn_wmma_f32_16x16x4_f32(
                  false, a, false, bw, (short)0, acc, false, false);
    }

    // D column n = lane&15 all identical; lanes 0 and 16 hold rows {0..7}/{8..15}
    if ((lane & 15) == 0) {
        float* o = out + b * 196 + g * 49 + s + 8 * half;
        #pragma unroll
        for (int i = 0; i < 8; ++i) o[i] = acc[i];
    }
}

extern "C" void kernel_launch(void* const* d_in, const int* in_sizes, int n_in,
                              void* d_out, int out_size, void* d_ws, size_t ws_size,
                              hipStream_t stream) {
    const float* x    = (const float*)d_in[0];
    const float* W    = (const float*)d_in[1];
    const float* bias = (const float*)d_in[2];
    float* out        = (float*)d_out;

    const int B = in_sizes[0] / 150528;        // 512
    dim3 grid(B * 4);
    dim3 block(256);
    size_t shmem = (size_t)LDS_FLOATS * sizeof(float);   // ~151 KB -> 2 WGs/WGP
    hipLaunchKernelGGL(patch_scorer_wmma, grid, block, shmem, stream, x, W, bias, out);
}